// jslab_78554951844272
// MI455X (gfx1250) — compile-verified
//
#include <hip/hip_runtime.h>

// CDNA5 wave32 WMMA vector types
typedef __attribute__((ext_vector_type(2))) float v2f;   // 16x4 / 4x16 f32 operand: 2 VGPRs
typedef __attribute__((ext_vector_type(8))) float v8f;   // 16x16 f32 accumulator: 8 VGPRs

__device__ __forceinline__ void cmul(float ar, float ai, float br, float bi,
                                     float& cr, float& ci) {
    cr = ar * br - ai * bi;
    ci = ar * bi + ai * br;
}

// a^p by repeated complex multiply (p <= 16, matches the serial product chain)
__device__ __forceinline__ void cpow(float ar, float ai, int p, float& r, float& i) {
    float xr = 1.f, xi = 0.f;
    for (int q = 0; q < p; ++q) { float tr, ti; cmul(xr, xi, ar, ai, tr, ti); xr = tr; xi = ti; }
    r = xr; i = xi;
}

// ---------------------------------------------------------------------------
// Pass 1: per wave, one tile of 16 chunks x 16 steps.
//   S[c,k] = sum_{j<=k} a^{k-j} * b_{16c+j}   via  B_forcing(16x16) x M^T(16x16)
//   complex -> 4 real f32 GEMMs -> 16x v_wmma_f32_16x16x4_f32 (K split 4x4).
// S written directly into d_out (U=Re, V=Im interleaved).
// ---------------------------------------------------------------------------
__global__ void slab_pass1(const float* __restrict__ pk, const float* __restrict__ TAx,
                           const float* __restrict__ TAy, const float* __restrict__ fc,
                           const int* __restrict__ p_dt, const int* __restrict__ p_dtf,
                           float* __restrict__ out, int nsteps, int nchunks, int ntiles,
                           int nt)
{
    int gid  = blockIdx.x * blockDim.x + threadIdx.x;
    int tile = gid >> 5;
    int lane = gid & 31;
    if (tile >= ntiles) return;           // whole wave exits -> EXEC all-1s for WMMA

    float dt      = (float)p_dt[0];
    int   nsub    = p_dtf[0] / p_dt[0];
    float invnsub = 1.f / (float)nsub;
    float K0   = __expf(pk[0]);
    float K1   = __expf(pk[1]);
    float f    = fc[0];
    float ar   = 1.f - dt * K1;           // a = (1 - dt*K1) - i*dt*f
    float ai   = -dt * f;
    float dtK0 = dt * K0;

    int half = lane >> 4;                 // ISA A-layout: lanes 16-31 carry K=2,3
    int r    = lane & 15;                 // A row (chunk-in-tile) == B/D column (k)
    int c0   = tile * 16;

    v2f Are[4], Aim[4], Mre[4], Mim[4], Mneg[4];
    #pragma unroll
    for (int s = 0; s < 4; ++s) {
        #pragma unroll
        for (int e = 0; e < 2; ++e) {
            int j = 4 * s + 2 * half + e;              // K index (step-in-chunk)
            // ---- A operand: b for chunk row r at step j (forcing interpolation)
            int  c  = c0 + r;
            long n  = (long)16 * c + j;
            int itf0 = (int)(n / nsub);
            float aa = (float)(n - (long)itf0 * nsub) * invnsub;
            int i0 = itf0     < nt - 1 ? itf0     : nt - 1;
            int i1 = itf0 + 1 < nt - 1 ? itf0 + 1 : nt - 1;
            float tax = (1.f - aa) * TAx[i0] + aa * TAx[i1];
            float tay = (1.f - aa) * TAy[i0] + aa * TAy[i1];
            float valid = (c < nchunks && n < (long)nsteps) ? 1.f : 0.f;
            Are[s][e] = valid * dtK0 * tax;
            Aim[s][e] = valid * dtK0 * tay;
            // ---- B operand: M^T[j][k] = a^(k-j) (k>=j), k = r
            int p = r - j;
            float pr, pi;
            cpow(ar, ai, p < 0 ? 0 : p, pr, pi);
            float m = (p >= 0) ? 1.f : 0.f;
            Mre[s][e]  =  m * pr;
            Mim[s][e]  =  m * pi;
            Mneg[s][e] = -m * pi;                      // f32 WMMA has no A/B neg
        }
    }

    v8f Sre = {}; v8f Sim = {};
    #pragma unroll
    for (int s = 0; s < 4; ++s)
        Sre = __builtin_amdgcn_wmma_f32_16x16x4_f32(false, Are[s], false, Mre[s],
                                                    (short)0, Sre, false, false);
    #pragma unroll
    for (int s = 0; s < 4; ++s)
        Sre = __builtin_amdgcn_wmma_f32_16x16x4_f32(false, Aim[s], false, Mneg[s],
                                                    (short)0, Sre, false, false);
    #pragma unroll
    for (int s = 0; s < 4; ++s)
        Sim = __builtin_amdgcn_wmma_f32_16x16x4_f32(false, Are[s], false, Mim[s],
                                                    (short)0, Sim, false, false);
    #pragma unroll
    for (int s = 0; s < 4; ++s)
        Sim = __builtin_amdgcn_wmma_f32_16x16x4_f32(false, Aim[s], false, Mre[s],
                                                    (short)0, Sim, false, false);

    // D layout: vgpr e -> M = e + 8*half, lane%16 -> N = k. Coalesced b64 stores.
    #pragma unroll
    for (int e = 0; e < 8; ++e) {
        int  row = e + 8 * half;
        int  c   = c0 + row;
        long n   = (long)16 * c + r;
        if (c < nchunks && n < (long)nsteps) {
            float2 v; v.x = Sre[e]; v.y = Sim[e];
            *reinterpret_cast<float2*>(out + 2 * n) = v;
        }
    }
}

// ---------------------------------------------------------------------------
// Pass 2: single block. Scan the per-chunk transfer functions
//   W_start(c+1) = a^16 * W_start(c) + S[c,15]
// thread-local compose + Hillis-Steele LDS scan of (A,B) function pairs.
// Writes W_start per chunk into d_ws (float2).
// ---------------------------------------------------------------------------
__global__ void slab_pass2(const float* __restrict__ pk, const float* __restrict__ fc,
                           const int* __restrict__ p_dt,
                           const float* __restrict__ out, float2* __restrict__ ws,
                           int nsteps, int nchunks)
{
    __shared__ float4 sh[1024];
    int t = threadIdx.x;
    float dt = (float)p_dt[0];
    float K1 = __expf(pk[1]);
    float f  = fc[0];
    float ar = 1.f - dt * K1, ai = -dt * f;
    float a16r, a16i; cpow(ar, ai, 16, a16r, a16i);

    int cpt = (nchunks + 1023) >> 10;
    int cb  = t * cpt;
    int ce  = min(cb + cpt, nchunks);

    float Ar = 1.f, Ai = 0.f, Br = 0.f, Bi = 0.f;   // g(W) = A*W + B
    for (int c = cb; c < ce; ++c) {
        long n15 = (long)16 * c + 15;
        float gr = 0.f, gi = 0.f;
        if (n15 < (long)nsteps) { gr = out[2 * n15]; gi = out[2 * n15 + 1]; }
        float tr, ti;
        cmul(a16r, a16i, Ar, Ai, tr, ti); Ar = tr; Ai = ti;
        cmul(a16r, a16i, Br, Bi, tr, ti); Br = tr + gr; Bi = ti + gi;
    }
    sh[t] = make_float4(Ar, Ai, Br, Bi);
    __syncthreads();
    for (int off = 1; off < 1024; off <<= 1) {
        float4 v = sh[t];
        if (t >= off) {
            float4 p = sh[t - off];     // earlier segment; result = v ∘ p
            float nar, nai, tr, ti;
            cmul(v.x, v.y, p.x, p.y, nar, nai);
            cmul(v.x, v.y, p.z, p.w, tr, ti);
            v = make_float4(nar, nai, tr + v.z, ti + v.w);
        }
        __syncthreads();
        sh[t] = v;
        __syncthreads();
    }
    // exclusive prefix applied to W=0 -> B component
    float Wr = 0.f, Wi = 0.f;
    if (t > 0) { float4 p = sh[t - 1]; Wr = p.z; Wi = p.w; }
    for (int c = cb; c < ce; ++c) {
        ws[c] = make_float2(Wr, Wi);
        long n15 = (long)16 * c + 15;
        float gr = 0.f, gi = 0.f;
        if (n15 < (long)nsteps) { gr = out[2 * n15]; gi = out[2 * n15 + 1]; }
        float tr, ti;
        cmul(a16r, a16i, Wr, Wi, tr, ti);
        Wr = tr + gr; Wi = ti + gi;
    }
}

// ---------------------------------------------------------------------------
// Pass 3: ys[16c+k] = a^(k+1)*W_start(c) + S[c,k]  (in-place add into d_out)
// ---------------------------------------------------------------------------
__global__ void slab_pass3(const float* __restrict__ pk, const float* __restrict__ fc,
                           const int* __restrict__ p_dt,
                           const float2* __restrict__ ws, float* __restrict__ out,
                           int nsteps)
{
    long n = (long)blockIdx.x * blockDim.x + threadIdx.x;
    if (n >= (long)nsteps) return;
    float dt = (float)p_dt[0];
    float K1 = __expf(pk[1]);
    float f  = fc[0];
    float ar = 1.f - dt * K1, ai = -dt * f;
    int c = (int)(n >> 4);
    int k = (int)(n & 15);
    float2 W = ws[c];
    float mr, mi; cpow(ar, ai, k + 1, mr, mi);
    out[2 * n]     += mr * W.x - mi * W.y;
    out[2 * n + 1] += mr * W.y + mi * W.x;
}

extern "C" void kernel_launch(void* const* d_in, const int* in_sizes, int n_in,
                              void* d_out, int out_size, void* d_ws, size_t ws_size,
                              hipStream_t stream)
{
    const float* pk    = (const float*)d_in[0];
    const float* TAx   = (const float*)d_in[1];
    const float* TAy   = (const float*)d_in[2];
    const float* fc    = (const float*)d_in[3];
    const int*   p_dt  = (const int*)d_in[6];
    const int*   p_dtf = (const int*)d_in[7];
    (void)n_in; (void)ws_size;

    int nt      = in_sizes[1];
    int nsteps  = out_size / 2;
    int nchunks = (nsteps + 15) / 16;
    int ntiles  = (nchunks + 15) / 16;
    float*  out = (float*)d_out;
    float2* ws  = (float2*)d_ws;

    int threads1 = ntiles * 32;
    int blocks1  = (threads1 + 255) / 256;
    slab_pass1<<<blocks1, 256, 0, stream>>>(pk, TAx, TAy, fc, p_dt, p_dtf,
                                            out, nsteps, nchunks, ntiles, nt);
    slab_pass2<<<1, 1024, 0, stream>>>(pk, fc, p_dt, out, ws, nsteps, nchunks);
    int blocks3 = (nsteps + 255) / 256;
    slab_pass3<<<blocks3, 256, 0, stream>>>(pk, fc, p_dt, ws, out, nsteps);
}